// VectorQuantizer_9131100471899
// MI455X (gfx1250) — compile-verified
//
#include <hip/hip_runtime.h>
#include <hip/hip_bf16.h>

// ---------------------------------------------------------------------------
// VQ-VAE vector quantizer for MI455X (gfx1250, wave32).
//   inputs:    (8, 64, 8, 32, 32) f32   -> N = 65536 vectors of dim 64
//   embedding: (1024, 64) f32
// Outputs (concatenated in d_out, float):
//   quant_st (4194304) | loss (1) | idx (65536, stored as float)
// Distance GEMM via V_WMMA_F32_16X16X4_F32 (f32 kept for exact argmin).
// Codebook staged into LDS with double-buffered GLOBAL_LOAD_ASYNC_TO_LDS_B128
// so the LDS fill overlaps the WMMA phase (ASYNCcnt + barriers for sync).
// ---------------------------------------------------------------------------

typedef float v2f __attribute__((ext_vector_type(2)));
typedef float v8f __attribute__((ext_vector_type(8)));
typedef int   v4i_ __attribute__((vector_size(16)));   // builtin's b128 unit

#define AS1 __attribute__((address_space(1)))
#define AS3 __attribute__((address_space(3)))

#define K_CODES   1024
#define DIM       64
#define N_POS     65536      // 8 * 8*32*32
#define POS_PER_B 8192       // L*H*W = 8*32*32
#define B_STRIDE  524288     // C * POS_PER_B
#define SLICE     64         // codes per LDS slice (double buffered)
#define NSLICES   (K_CODES / SLICE)          // 16
#define ES_STRIDE 68         // padded row stride: 272B rows -> 16B aligned for
                             // async b128 writes AND conflict-free ds_load_b64
#define ES_BUF    (SLICE * ES_STRIDE)        // floats per buffer
#define QUANT_ELEMS 4194304

// --------------------------- pass 1: ||e_k||^2 ------------------------------
__global__ void vq_enorm(const float* __restrict__ emb, float* __restrict__ enorm) {
  int k = blockIdx.x * blockDim.x + threadIdx.x;
  if (k >= K_CODES) return;
  const float4* e4 = (const float4*)(emb + (size_t)k * DIM);
  float s = 0.f;
#pragma unroll
  for (int i = 0; i < DIM / 4; ++i) {
    float4 v = e4[i];
    s += v.x * v.x + v.y * v.y + v.z * v.z + v.w * v.w;
  }
  enorm[k] = s;
}

// --------------------------- pass 2: WMMA argmin ----------------------------
// Block = 1024 threads = 32 waves. Each wave owns one 16-row tile of X.
// Codebook streamed through 2 LDS buffers of 64 codes via async LDS loads.
__launch_bounds__(1024, 1)
__global__ void vq_argmin(const float* __restrict__ inp,
                          const float* __restrict__ emb,
                          const float* __restrict__ enorm,
                          float* __restrict__ idx_out) {
  __shared__ float Es[2 * ES_BUF];   // 2 x 17408 B
  __shared__ float enAll[K_CODES];   // 4 KB: all ||e||^2 staged once

  const int tid  = threadIdx.x;
  const int lane = tid & 31;
  const int wave = tid >> 5;
  const int m    = lane & 15;   // A row / D column index
  const int hi   = lane >> 4;   // half-wave select

  const int tile = blockIdx.x * 32 + wave;  // 4096 tiles total
  const int n0   = tile * 16;               // never crosses a batch block
  const int b    = n0 >> 13;
  const int r0   = n0 & (POS_PER_B - 1);

  // Per-thread async slice-load slot: one 16B row segment per lane.
  const int a_code = tid >> 4;   // 0..63
  const int a_seg  = tid & 15;   // 0..15
  const float* gsrc0 = emb + (size_t)a_code * DIM + (size_t)a_seg * 4;
  float* ldst0 = &Es[a_code * ES_STRIDE + a_seg * 4];
  float* ldst1 = ldst0 + ES_BUF;

  enAll[tid] = enorm[tid];       // 1024 threads, one value each

  // Kick off slice 0 into buffer 0 (ASYNCcnt tracked, no VGPR round-trip).
  __builtin_amdgcn_global_load_async_to_lds_b128(
      (AS1 v4i_*)gsrc0, (AS3 v4i_*)ldst0, 0, 0);

  // Preload A tile (16 rows x 64 ch) into 16 v2f registers, pre-scaled by -2.
  // Per WMMA step s: lane holds A[m][c0], A[m][c0+1], c0 = 4s + 2*hi.
  const float* abase = inp + (size_t)b * B_STRIDE + r0 + m;
  v2f a[16];
#pragma unroll
  for (int s = 0; s < 16; ++s) {
    const int c0 = 4 * s + 2 * hi;
    a[s][0] = -2.0f * abase[(size_t)c0 * POS_PER_B];
    a[s][1] = -2.0f * abase[(size_t)(c0 + 1) * POS_PER_B];
  }

  float best[8];
  int   bidx[8];
#pragma unroll
  for (int j = 0; j < 8; ++j) { best[j] = 3.0e38f; bidx[j] = 0; }

  for (int sl = 0; sl < NSLICES; ++sl) {
    const int k0 = sl * SLICE;
    // Issue the NEXT slice into the other buffer, then wait only for the
    // CURRENT slice (async loads complete in order -> asynccnt <= 1).
    if (sl + 1 < NSLICES) {
      const float* g = gsrc0 + (size_t)(k0 + SLICE) * DIM;
      float* l = ((sl + 1) & 1) ? ldst1 : ldst0;
      __builtin_amdgcn_global_load_async_to_lds_b128(
          (AS1 v4i_*)g, (AS3 v4i_*)l, 0, 0);
      __builtin_amdgcn_s_wait_asynccnt(1);
    } else {
      __builtin_amdgcn_s_wait_asynccnt(0);
    }
    __syncthreads();   // all waves' portions of slice sl are in LDS

    const float* EsBuf = &Es[(sl & 1) * ES_BUF];
#pragma unroll
    for (int ch = 0; ch < SLICE / 16; ++ch) {
      const int cc = ch * 16;               // 16-code chunk within slice
      v8f acc = {};
      // B step s: lane reads E[code = cc+m][c0..c0+1]; stride-68 rows put
      // half hi=0 on banks {0,1} mod 4 and hi=1 on {2,3} mod 4 -> no conflicts.
      const float* bp = &EsBuf[(cc + m) * ES_STRIDE + 2 * hi];
#pragma unroll
      for (int s = 0; s < 16; ++s) {
        v2f bv = *(const v2f*)(bp + 4 * s);
        acc = __builtin_amdgcn_wmma_f32_16x16x4_f32(
            false, a[s], false, bv, (short)0, acc, false, false);
      }
      // acc[j] = -2 * dot(x_{row j+8*hi}, e_{k0+cc+m}); score adds ||e||^2.
      const float ev   = enAll[k0 + cc + m];
      const int   code = k0 + cc + m;
#pragma unroll
      for (int j = 0; j < 8; ++j) {
        const float sc = ev + acc[j];
        if (sc < best[j]) { best[j] = sc; bidx[j] = code; }
      }
    }
    __syncthreads();   // everyone done reading this buffer before next refill
  }

  // Min-reduce the 16 code-columns held across each 16-lane half (wave32:
  // xor masks 1,2,4,8 never cross the half boundary). Ties -> lower index.
#pragma unroll
  for (int off = 8; off >= 1; off >>= 1) {
#pragma unroll
    for (int j = 0; j < 8; ++j) {
      const float os = __shfl_xor(best[j], off, 32);
      const int   oi = __shfl_xor(bidx[j], off, 32);
      if (os < best[j] || (os == best[j] && oi < bidx[j])) {
        best[j] = os; bidx[j] = oi;
      }
    }
  }
  if (m == 0) {  // lane 0 -> rows 0..7, lane 16 -> rows 8..15
#pragma unroll
    for (int j = 0; j < 8; ++j)
      idx_out[n0 + hi * 8 + j] = (float)bidx[j];
  }
}

// ------------------- pass 3: gather, scatter, loss reduce -------------------
__global__ void vq_gather_loss(const float* __restrict__ inp,
                               const float* __restrict__ emb,
                               const float* __restrict__ idx_f,
                               float* __restrict__ quant,
                               float* __restrict__ loss) {
  __shared__ int   sidx[64];
  __shared__ float red[256];
  const int tid = threadIdx.x;
  const int n0  = blockIdx.x * 64;
  if (tid < 64) sidx[tid] = (int)idx_f[n0 + tid];
  __syncthreads();

  const int p     = tid & 63;   // position within block's 64-pos strip
  const int cbase = tid >> 6;   // 0..3
  const int n     = n0 + p;
  const int b     = n >> 13;
  const int r     = n & (POS_PER_B - 1);
  const size_t base = (size_t)b * B_STRIDE + r;
  const float* erow = emb + (size_t)sidx[p] * DIM;

  float lsum = 0.f;
#pragma unroll
  for (int i = 0; i < 16; ++i) {
    const int c = i * 4 + cbase;
    const size_t off = base + (size_t)c * POS_PER_B;  // coalesced over p
    const float q = erow[c];
    const float x = inp[off];
    const float d = q - x;
    lsum += d * d;
    quant[off] = q;   // quant_st == quant in value (straight-through)
  }
  red[tid] = lsum;
  __syncthreads();
  for (int s = 128; s > 0; s >>= 1) {
    if (tid < s) red[tid] += red[tid + s];
    __syncthreads();
  }
  // loss = q_lat + 0.25*e_lat, and both equal mean((quant-x)^2) in value.
  if (tid == 0) atomicAdd(loss, red[0] * (1.25f / (float)QUANT_ELEMS));
}

// ---------------------------------------------------------------------------
extern "C" void kernel_launch(void* const* d_in, const int* in_sizes, int n_in,
                              void* d_out, int out_size, void* d_ws, size_t ws_size,
                              hipStream_t stream) {
  const float* inp = (const float*)d_in[0];
  const float* emb = (const float*)d_in[1];
  float* out   = (float*)d_out;
  float* quant = out;                    // [0, 4194304)
  float* loss  = out + QUANT_ELEMS;      // [4194304]
  float* idx_f = out + QUANT_ELEMS + 1;  // [4194305, +65536)
  float* enorm = (float*)d_ws;           // 1024 floats of scratch

  vq_enorm<<<K_CODES / 256, 256, 0, stream>>>(emb, enorm);
  vq_argmin<<<(N_POS / 16) / 32, 1024, 0, stream>>>(inp, emb, enorm, idx_f);
  (void)hipMemsetAsync(loss, 0, sizeof(float), stream);
  vq_gather_loss<<<N_POS / 64, 256, 0, stream>>>(inp, emb, idx_f, quant, loss);
}